// LDA_60859686584682
// MI455X (gfx1250) — compile-verified
//
#include <hip/hip_runtime.h>
#include <hip/hip_bf16.h>

#define N_ROWS 131072
#define DIMS   512
#define NCLS   100
#define LAMB   0.001f
#define SPLITK 32
#define LDW    132   // 128 + 4 pad, floats
#define TILEF  (32 * LDW)   // floats per tile buffer

typedef __attribute__((ext_vector_type(2))) float v2f;
typedef __attribute__((ext_vector_type(8))) float v8f;

// ---------------------------------------------------------------- zero
__global__ void zero_kernel(float* p, int n) {
    int i = blockIdx.x * blockDim.x + threadIdx.x;
    if (i < n) p[i] = 0.0f;
}

// ---------------------------------------------------------------- stats
// Per block: 2048 rows. Dynamic LDS: cls[100*512] + ccnt[128].
// Column d is owned by thread (d % 256), so no LDS atomics needed.
__global__ void stats_kernel(const float* __restrict__ X, const int* __restrict__ y,
                             float* __restrict__ colsum_g, float* __restrict__ counts_g,
                             float* __restrict__ classsum_g) {
    extern __shared__ float smem[];
    float* cls  = smem;            // 51200
    float* ccnt = smem + 51200;    // 128
    const int tid = threadIdx.x;   // 256 threads

    for (int t = tid; t < 51200 + 128; t += 256) smem[t] = 0.0f;
    __syncthreads();

    float c0 = 0.0f, c1 = 0.0f;
    const int base = blockIdx.x * 2048;
    for (int rr = 0; rr < 2048; ++rr) {
        const int n = base + rr;
        const int c = y[n];                       // uniform -> scalar load
        const float x0 = X[(size_t)n * DIMS + tid];
        const float x1 = X[(size_t)n * DIMS + tid + 256];
        c0 += x0; c1 += x1;
        float* row = cls + c * DIMS;
        row[tid]       += x0;
        row[tid + 256] += x1;
        if (tid == 0) ccnt[c] += 1.0f;
    }
    __syncthreads();

    atomicAdd(&colsum_g[tid],       c0);
    atomicAdd(&colsum_g[tid + 256], c1);
    for (int t = tid; t < NCLS * DIMS; t += 256) atomicAdd(&classsum_g[t], cls[t]);
    if (tid < NCLS) atomicAdd(&counts_g[tid], ccnt[tid]);
}

// ---------------------------------------------------------------- finalize stats
__global__ void finalize_stats(const float* __restrict__ colsum, const float* __restrict__ counts,
                               const float* __restrict__ classsum,
                               float* __restrict__ meanv, float* __restrict__ cmean,
                               float* __restrict__ wc, float* __restrict__ mu_out) {
    const int d = threadIdx.x;                    // 512 threads
    if (blockIdx.x == 0) {
        const float m = colsum[d] * (1.0f / (float)N_ROWS);
        meanv[d] = m;
        mu_out[d] = m;
        if (d < NCLS) {
            const float cnt = counts[d];
            wc[d] = cnt / fmaxf(cnt - 1.0f, 1.0f) / (float)N_ROWS;
        }
    } else {
        const int c = blockIdx.x - 1;
        const float cnt = fmaxf(counts[c], 1.0f);
        cmean[c * DIMS + d] = classsum[c * DIMS + d] / cnt;
    }
}

// ---------------------------------------------------------------- per-row weight
__global__ void wrow_kernel(const int* __restrict__ y, const float* __restrict__ wc,
                            float* __restrict__ wrow) {
    const int n = blockIdx.x * blockDim.x + threadIdx.x;
    if (n < N_ROWS) wrow[n] = wc[y[n]];
}

// ---------------------------------------------------------------- async tile prefetch (global -> LDS DMA)
__device__ __forceinline__ void prefetch_tiles(const float* __restrict__ X,
                                               const float* __restrict__ wrow,
                                               float* tIb, float* tJb, float* wshb,
                                               int nb, int i0, int j0, int tid) {
    const unsigned ldsI = (unsigned)(uintptr_t)tIb;
    const unsigned ldsJ = (unsigned)(uintptr_t)tJb;
    const unsigned long long xbase = (unsigned long long)(uintptr_t)X;
    // 32 rows x 128 cols per tile = 1024 float4; 256 threads -> 4 each per tile
    for (int t = tid; t < 1024; t += 256) {
        const int row = t >> 5;             // 0..31
        const int c4  = (t & 31) << 2;      // 0,4,...,124
        const unsigned loff = (unsigned)(row * LDW + c4) * 4u;
        const unsigned ga = (unsigned)(((nb + row) * DIMS + i0 + c4) * 4);
        const unsigned la = ldsI + loff;
        asm volatile("global_load_async_to_lds_b128 %0, %1, %2"
                     :: "v"(la), "v"(ga), "s"(xbase) : "memory");
        const unsigned gb = (unsigned)(((nb + row) * DIMS + j0 + c4) * 4);
        const unsigned lb = ldsJ + loff;
        asm volatile("global_load_async_to_lds_b128 %0, %1, %2"
                     :: "v"(lb), "v"(gb), "s"(xbase) : "memory");
    }
    if (tid < 32) wshb[tid] = wrow[nb + tid];
}

__device__ __forceinline__ void wait_async0() {
    asm volatile("s_wait_asynccnt 0x0" ::: "memory");
}

// ---------------------------------------------------------------- dual Gram via f32 WMMA
// grid (4,4,SPLITK), 256 threads. Block computes 128x128 tile of G and Gw over
// its N-slice. Wave w owns rows w*16..+16 and all 128 cols (8 col subtiles),
// each with a plain (G) and weighted (Gw) f32 accumulator.
// Dynamic LDS: double-buffered tI[2][32][LDW], tJ[2][32][LDW], wsh[2][32].
__global__ void __launch_bounds__(256)
gram_kernel(const float* __restrict__ X, const float* __restrict__ wrow,
            float* __restrict__ G, float* __restrict__ Gw) {
    extern __shared__ float smem[];
    // layout (floats): tI[2][TILEF] | tJ[2][TILEF] | wsh[2][32]
    float* tI_base  = smem;
    float* tJ_base  = smem + 2 * TILEF;
    float* wsh_base = smem + 4 * TILEF;

    const int tid  = threadIdx.x;
    const int lane = tid & 31;
    const int wave = tid >> 5;
    const int w16  = wave * 16;         // row subtile of this wave
    const int m    = lane & 15;
    const int kh   = lane >> 4;         // K half (A/B fragment layout)

    const int i0 = blockIdx.x * 128;
    const int j0 = blockIdx.y * 128;
    const int nbeg = blockIdx.z * (N_ROWS / SPLITK);
    const int nend = nbeg + (N_ROWS / SPLITK);

    v8f accG[8] = {};
    v8f accW[8] = {};

    int buf = 0;
    prefetch_tiles(X, wrow, tI_base, tJ_base, wsh_base, nbeg, i0, j0, tid);
    wait_async0();
    __syncthreads();

    for (int nb = nbeg; nb < nend; nb += 32) {
        const int nxt = nb + 32;
        const int nbuf = buf ^ 1;
        if (nxt < nend)
            prefetch_tiles(X, wrow,
                           tI_base + nbuf * TILEF,
                           tJ_base + nbuf * TILEF,
                           wsh_base + nbuf * 32,
                           nxt, i0, j0, tid);

        const float* tIb  = tI_base + buf * TILEF;
        const float* tJb  = tJ_base + buf * TILEF;
        const float* wshb = wsh_base + buf * 32;

#pragma unroll
        for (int k = 0; k < 32; k += 4) {
            const int k0 = k + kh * 2;
            const float* rI0 = tIb + k0 * LDW;
            const float* rJ0 = tJb + k0 * LDW;
            v2f a;  a.x = rI0[w16 + m];  a.y = rI0[LDW + w16 + m];
            const float wk0 = wshb[k0], wk1 = wshb[k0 + 1];
#pragma unroll
            for (int s = 0; s < 8; ++s) {
                v2f b;  b.x  = rJ0[s * 16 + m];       b.y  = rJ0[LDW + s * 16 + m];
                v2f bw; bw.x = b.x * wk0;             bw.y = b.y * wk1;
                accG[s] = __builtin_amdgcn_wmma_f32_16x16x4_f32(false, a, false, b,
                                                                (short)0, accG[s], false, false);
                accW[s] = __builtin_amdgcn_wmma_f32_16x16x4_f32(false, a, false, bw,
                                                                (short)0, accW[s], false, false);
            }
        }

        wait_async0();      // our prefetch of next buffer has landed in LDS
        __syncthreads();    // everyone done computing on buf + next buffer ready
        buf ^= 1;
    }

    // C/D layout: VGPR v, lanes 0-15 -> M=v, lanes 16-31 -> M=v+8; N = lane&15
    const int gi_base = i0 + w16 + kh * 8;
#pragma unroll
    for (int s = 0; s < 8; ++s) {
        const int gj = j0 + s * 16 + m;
#pragma unroll
        for (int v = 0; v < 8; ++v) {
            const int gi = gi_base + v;
            atomicAdd(&G [gi * DIMS + gj], accG[s][v]);
            atomicAdd(&Gw[gi * DIMS + gj], accW[s][v]);
        }
    }
}

// ---------------------------------------------------------------- finalize Sw/Sb/St + build [Sw+lI | Sb]
__global__ void finalize_mats(const float* __restrict__ G, const float* __restrict__ Gw,
                              const float* __restrict__ meanv, const float* __restrict__ cmean,
                              const float* __restrict__ counts, const float* __restrict__ wc,
                              float* __restrict__ sw_out, float* __restrict__ sb_out,
                              float* __restrict__ st_out, float* __restrict__ M) {
    __shared__ float scnt[NCLS], swc[NCLS];
    if (threadIdx.x < NCLS) { scnt[threadIdx.x] = counts[threadIdx.x]; swc[threadIdx.x] = wc[threadIdx.x]; }
    __syncthreads();

    const int idx = blockIdx.x * blockDim.x + threadIdx.x;   // 512*512 total
    const int i = idx >> 9;
    const int j = idx & 511;
    const float mi = meanv[i], mj = meanv[j];

    const float st = (G[idx] - (float)N_ROWS * mi * mj) * (1.0f / (float)(N_ROWS - 1));

    float accSw = 0.0f, accSb = 0.0f;
    for (int c = 0; c < NCLS; ++c) {
        const float mci = cmean[c * DIMS + i];
        const float mcj = cmean[c * DIMS + j];
        accSw += swc[c] * scnt[c] * mci * mcj;
        accSb += (scnt[c] * (1.0f / (float)N_ROWS)) * (mci - mi) * (mcj - mj);
    }
    const float sw = Gw[idx] - accSw;

    sw_out[idx] = sw;
    sb_out[idx] = accSb;
    st_out[idx] = st;
    M[(size_t)i * 1024 + j]       = sw + ((i == j) ? LAMB : 0.0f);
    M[(size_t)i * 1024 + 512 + j] = accSb;
}

// ---------------------------------------------------------------- Gauss-Jordan solve (SPD), symmetrize
__global__ void __launch_bounds__(1024)
solve_kernel(float* __restrict__ M, float* __restrict__ temp_out) {
    __shared__ float prow[1024];
    __shared__ float pcol[512];
    const int tid = threadIdx.x;   // 1024 threads, thread owns column tid

    for (int p = 0; p < 512; ++p) {
        const float inv = 1.0f / M[(size_t)p * 1024 + p];
        prow[tid] = M[(size_t)p * 1024 + tid] * inv;
        if (tid < 512) pcol[tid] = (tid == p) ? 0.0f : M[(size_t)tid * 1024 + p];
        __syncthreads();

        M[(size_t)p * 1024 + tid] = prow[tid];
        const float pr = prow[tid];
        for (int r = 0; r < 512; ++r) {
            if (r != p) M[(size_t)r * 1024 + tid] -= pcol[r] * pr;
        }
        __syncthreads();
    }

    // temp = (Ainv*B + (Ainv*B)^T)/2 ; solution sits in columns 512..1023
    for (int t = tid; t < 512 * 512; t += 1024) {
        const int i = t >> 9;
        const int j = t & 511;
        temp_out[t] = 0.5f * (M[(size_t)i * 1024 + 512 + j] + M[(size_t)j * 1024 + 512 + i]);
    }
}

// ---------------------------------------------------------------- launch
extern "C" void kernel_launch(void* const* d_in, const int* in_sizes, int n_in,
                              void* d_out, int out_size, void* d_ws, size_t ws_size,
                              hipStream_t stream) {
    (void)in_sizes; (void)n_in; (void)out_size; (void)ws_size;
    const float* X = (const float*)d_in[0];
    const int*   y = (const int*)d_in[1];
    float* out = (float*)d_out;
    float* ws  = (float*)d_ws;

    // workspace layout (floats)
    float* colsum   = ws;               // 512
    float* counts   = ws + 512;         // 128
    float* classsum = ws + 640;         // 51200
    float* meanv    = ws + 51840;       // 512
    float* cmean    = ws + 52352;       // 51200
    float* wc       = ws + 103552;      // 128
    float* wrow     = ws + 103680;      // 131072
    float* G        = ws + 234752;      // 262144
    float* Gw       = ws + 496896;      // 262144
    float* M        = ws + 759040;      // 512*1024

    // output layout: mu(512) | temp | Sw | Sb | St
    float* mu_out   = out;
    float* temp_out = out + 512;
    float* sw_out   = temp_out + 262144;
    float* sb_out   = sw_out + 262144;
    float* st_out   = sb_out + 262144;

    zero_kernel<<<(51840 + 255) / 256, 256, 0, stream>>>(ws, 51840);        // stats region
    zero_kernel<<<(524288 + 255) / 256, 256, 0, stream>>>(G, 524288);       // G + Gw

    stats_kernel<<<64, 256, (51200 + 128) * sizeof(float), stream>>>(X, y, colsum, counts, classsum);
    finalize_stats<<<NCLS + 1, 512, 0, stream>>>(colsum, counts, classsum, meanv, cmean, wc, mu_out);
    wrow_kernel<<<(N_ROWS + 255) / 256, 256, 0, stream>>>(y, wc, wrow);

    dim3 grid(4, 4, SPLITK);
    const size_t gram_lds = (size_t)(4 * TILEF + 64) * sizeof(float);       // ~66 KB dynamic LDS
    gram_kernel<<<grid, 256, gram_lds, stream>>>(X, wrow, G, Gw);

    finalize_mats<<<(512 * 512) / 256, 256, 0, stream>>>(G, Gw, meanv, cmean, counts, wc,
                                                         sw_out, sb_out, st_out, M);
    solve_kernel<<<1, 1024, 0, stream>>>(M, temp_out);
}